// GIN_2276332667486
// MI455X (gfx1250) — compile-verified
//
#include <hip/hip_runtime.h>
#include <math.h>

#define NNODES 100000
#define NEDGES 1600000
#define NGRAPH 256
#define HDIM   64
#define TS 68   // LDS tile row stride (floats): 16B-aligned rows, conflict-reducing

typedef float v2f __attribute__((ext_vector_type(2)));
typedef float v8f __attribute__((ext_vector_type(8)));

__device__ __forceinline__ void atomAddF(float* p, float v) {
    __hip_atomic_fetch_add(p, v, __ATOMIC_RELAXED, __HIP_MEMORY_SCOPE_AGENT);
}

// ---------------- utility: float4 copy (agg = h_prev, then scatter adds) ---
__global__ void copy_kernel(const float* __restrict__ src, float* __restrict__ dst, int n4) {
    int i = blockIdx.x * blockDim.x + threadIdx.x;
    if (i < n4) ((float4*)dst)[i] = ((const float4*)src)[i];
}

// ---------------- edge scatter: agg[dst] += h[src] ------------------------
// thread = (edge, 4-feature chunk); 16 threads per edge; edge uniform per 16 lanes
__global__ void scatter_kernel(const float* __restrict__ h, const int* __restrict__ ei,
                               float* __restrict__ agg, int nThreads) {
    int idx = blockIdx.x * blockDim.x + threadIdx.x;
    if (idx >= nThreads) return;
    int e  = idx >> 4;
    int c4 = (idx & 15) << 2;
    int s = ei[e];
    int d = ei[NEDGES + e];
    float4 val = *(const float4*)(h + (size_t)s * HDIM + c4);
    float* p = agg + (size_t)d * HDIM + c4;
    atomAddF(p + 0, val.x); atomAddF(p + 1, val.y);
    atomAddF(p + 2, val.z); atomAddF(p + 3, val.w);
}

// ---------------- fused MLP: out = relu(relu(bn(agg@w1+b1))@w2+b2) --------------
// one wave per 16-row tile; f32 WMMA 16x16x4.
// Weights pre-swizzled in LDS to B-fragment order: [kb][nt][lane][2] so each
// lane's B operand is one contiguous 8B ds_load_b64 (no VALU re-pack).
__global__ __launch_bounds__(256)
void gin_mlp_kernel(const float* __restrict__ in, float* __restrict__ out,
                    const float* __restrict__ w1, const float* __restrict__ b1,
                    const float* __restrict__ g,  const float* __restrict__ be,
                    const float* __restrict__ m,  const float* __restrict__ v,
                    const float* __restrict__ w2, const float* __restrict__ b2,
                    int nTiles)
{
    __shared__ __align__(16) float sWf1[4096];      // 16 kb * 4 nt * 32 lane * 2
    __shared__ __align__(16) float sWf2[4096];
    __shared__ __align__(16) float sT[8][16 * TS];

    // cooperative weight staging, swizzled to fragment order (one-time, L2-hot)
    for (int i = threadIdx.x; i < 4096; i += 256) {
        int kb   = i >> 8;
        int rem  = i & 255;
        int nt   = rem >> 6;
        int ln   = (rem >> 1) & 31;
        int half = rem & 1;
        int mr   = ln & 15;
        int kh2  = (ln >> 4) << 1;
        int k = (kb << 2) + kh2 + half;
        int n = (nt << 4) + mr;
        sWf1[i] = w1[k * HDIM + n];
        sWf2[i] = w2[k * HDIM + n];
    }
    __syncthreads();

    const int wave = threadIdx.x >> 5;
    const int lane = threadIdx.x & 31;
    const int tile = blockIdx.x * 8 + wave;
    if (tile >= nTiles) return;            // wave-uniform: EXEC stays all-ones

    float* tb = sT[wave];
    const int mrow  = lane & 15;           // M (A/C) or N (B/C) index
    const int kh2   = (lane >> 4) << 1;    // K sub-offset: 0 (lanes 0-15) / 2 (16-31)
    const int half8 = (lane >> 4) << 3;    // C/D row offset for upper lane half

    // coalesced tile load: global float4 -> LDS
    {
        const float* src = in + (size_t)tile * (16 * HDIM);
        for (int i = 0; i < 8; ++i) {
            int f = i * 32 + lane;                 // 0..255 float4 chunks
            int r = f >> 4, c4 = (f & 15) << 2;
            *(float4*)(tb + r * TS + c4) = *(const float4*)(src + r * HDIM + c4);
        }
    }

    // per-lane output-column constants (cn = mrow + n*16)
    float bias1[4], scl[4], sft[4], bias2[4];
    for (int n = 0; n < 4; ++n) {
        int cn = mrow + n * 16;
        float s = g[cn] * rsqrtf(v[cn] + 1e-5f);
        scl[n] = s;
        sft[n] = be[cn] - m[cn] * s;
        bias1[n] = b1[cn];
        bias2[n] = b2[cn];
    }

    const float* aBase  = tb + mrow * TS + kh2;      // A frag base (this lane)
    const float* bBase1 = sWf1 + (lane << 1);        // B frag base (this lane)
    const float* bBase2 = sWf2 + (lane << 1);

    const v8f vzero = {0.f,0.f,0.f,0.f,0.f,0.f,0.f,0.f};
    v8f acc[4];

    // ---- matmul 1: tile(16x64) @ w1(64x64) ----
    acc[0] = acc[1] = acc[2] = acc[3] = vzero;
    #pragma unroll
    for (int kb = 0; kb < 16; ++kb) {
        v2f a = *(const v2f*)(aBase + (kb << 2));
        #pragma unroll
        for (int nt = 0; nt < 4; ++nt) {
            v2f b = *(const v2f*)(bBase1 + (((kb << 2) + nt) << 6));
            acc[nt] = __builtin_amdgcn_wmma_f32_16x16x4_f32(
                false, a, false, b, (short)0, acc[nt], false, false);
        }
    }
    // bias + eval-BN + ReLU, re-layout C->A via LDS (same-wave LDS is in-order)
    #pragma unroll
    for (int n = 0; n < 4; ++n) {
        int cn = mrow + n * 16;
        #pragma unroll
        for (int r = 0; r < 8; ++r) {
            float val = scl[n] * (acc[n][r] + bias1[n]) + sft[n];
            tb[(r + half8) * TS + cn] = fmaxf(val, 0.0f);
        }
    }

    // ---- matmul 2: h(16x64) @ w2(64x64) ----
    acc[0] = acc[1] = acc[2] = acc[3] = vzero;
    #pragma unroll
    for (int kb = 0; kb < 16; ++kb) {
        v2f a = *(const v2f*)(aBase + (kb << 2));
        #pragma unroll
        for (int nt = 0; nt < 4; ++nt) {
            v2f b = *(const v2f*)(bBase2 + (((kb << 2) + nt) << 6));
            acc[nt] = __builtin_amdgcn_wmma_f32_16x16x4_f32(
                false, a, false, b, (short)0, acc[nt], false, false);
        }
    }
    #pragma unroll
    for (int n = 0; n < 4; ++n) {
        int cn = mrow + n * 16;
        #pragma unroll
        for (int r = 0; r < 8; ++r)
            tb[(r + half8) * TS + cn] = fmaxf(acc[n][r] + bias2[n], 0.0f);
    }

    // coalesced tile store: LDS -> global float4
    {
        float* dst = out + (size_t)tile * (16 * HDIM);
        for (int i = 0; i < 8; ++i) {
            int f = i * 32 + lane;
            int r = f >> 4, c4 = (f & 15) << 2;
            *(float4*)(dst + r * HDIM + c4) = *(const float4*)(tb + r * TS + c4);
        }
    }
}

// ---------------- global add pool per graph -------------------------------
__global__ void pool_kernel(const float* __restrict__ h, const int* __restrict__ batch,
                            float* __restrict__ pool, int off, int nThreads) {
    int idx = blockIdx.x * blockDim.x + threadIdx.x;
    if (idx >= nThreads) return;
    int node = idx >> 4;
    int c4   = (idx & 15) << 2;
    int gi = batch[node];
    float4 val = *(const float4*)(h + (size_t)node * HDIM + c4);
    float* p = pool + (size_t)gi * 192 + off + c4;
    atomAddF(p + 0, val.x); atomAddF(p + 1, val.y);
    atomAddF(p + 2, val.z); atomAddF(p + 3, val.w);
}

// ---------------- classifier: relu(p@l1+b)@l2+b -> log_softmax ------------
__global__ __launch_bounds__(192)
void cls_kernel(const float* __restrict__ pool,
                const float* __restrict__ l1w, const float* __restrict__ l1b,
                const float* __restrict__ l2w, const float* __restrict__ l2b,
                float* __restrict__ out) {
    __shared__ float sp[192];
    __shared__ float sh[192];
    __shared__ float sz[10];
    int gi = blockIdx.x, t = threadIdx.x;
    sp[t] = pool[(size_t)gi * 192 + t];
    __syncthreads();
    float acc = l1b[t];
    for (int k = 0; k < 192; ++k) acc = fmaf(sp[k], l1w[k * 192 + t], acc);
    sh[t] = fmaxf(acc, 0.0f);
    __syncthreads();
    if (t < 10) {
        float z = l2b[t];
        for (int k = 0; k < 192; ++k) z = fmaf(sh[k], l2w[k * 10 + t], z);
        sz[t] = z;
    }
    __syncthreads();
    if (t < 10) {
        float mx = sz[0];
        for (int c = 1; c < 10; ++c) mx = fmaxf(mx, sz[c]);
        float s = 0.0f;
        for (int c = 0; c < 10; ++c) s += expf(sz[c] - mx);
        out[(size_t)gi * 10 + t] = sz[t] - (mx + logf(s));
    }
}

extern "C" void kernel_launch(void* const* d_in, const int* in_sizes, int n_in,
                              void* d_out, int out_size, void* d_ws, size_t ws_size,
                              hipStream_t stream) {
    const float* x     = (const float*)d_in[0];
    const int*   ei    = (const int*)d_in[1];
    const int*   batch = (const int*)d_in[2];
    // per layer: w1,b1,w2,b2,g,be,m,v at d_in[3+8*l ..]
    const float* L[3][8];
    for (int l = 0; l < 3; ++l)
        for (int p = 0; p < 8; ++p)
            L[l][p] = (const float*)d_in[3 + l * 8 + p];
    const float* l1w = (const float*)d_in[27];
    const float* l1b = (const float*)d_in[28];
    const float* l2w = (const float*)d_in[29];
    const float* l2b = (const float*)d_in[30];
    float* out = (float*)d_out;

    float* bufA = (float*)d_ws;                       // N*64
    float* bufB = bufA + (size_t)NNODES * HDIM;       // N*64
    float* agg  = bufB + (size_t)NNODES * HDIM;       // N*64
    float* pool = agg  + (size_t)NNODES * HDIM;       // G*192

    hipMemsetAsync(pool, 0, (size_t)NGRAPH * 192 * sizeof(float), stream);

    const int nTiles    = NNODES / 16;                // 6250 (exact)
    const int mlpBlocks = (nTiles + 7) / 8;
    const int copyN4    = NNODES * 16;                // float4 chunks
    const int scatterT  = NEDGES * 16;
    const int poolT     = NNODES * 16;

    const float* hin = x;
    float* houts[3] = { bufA, bufB, bufA };           // h1 dead after layer-2 agg
    for (int l = 0; l < 3; ++l) {
        copy_kernel<<<(copyN4 + 255) / 256, 256, 0, stream>>>(hin, agg, copyN4);
        scatter_kernel<<<(scatterT + 255) / 256, 256, 0, stream>>>(hin, ei, agg, scatterT);
        gin_mlp_kernel<<<mlpBlocks, 256, 0, stream>>>(
            agg, houts[l],
            L[l][0], L[l][1],                    // w1, b1
            L[l][4], L[l][5], L[l][6], L[l][7],  // g, be, m, v
            L[l][2], L[l][3],                    // w2, b2
            nTiles);
        pool_kernel<<<(poolT + 255) / 256, 256, 0, stream>>>(houts[l], batch, pool, l * 64, poolT);
        hin = houts[l];
    }
    cls_kernel<<<NGRAPH, 192, 0, stream>>>(pool, l1w, l1b, l2w, l2b, out);
}